// SpectralGate1D_53180285059156
// MI455X (gfx1250) — compile-verified
//
#include <hip/hip_runtime.h>

// Problem constants (from reference setup_inputs)
#define B_DIM   8
#define T_DIM   4096
#define D_DIM   1024
#define K_FFT   64
#define NFREQ   33                 // K/2 + 1
#define TAIL_T0 (T_DIM - K_FFT)    // first nonzero timestep
#define DPB     8                  // d-channels per block (one per wave32)

typedef float v2f __attribute__((ext_vector_type(2)));
typedef float v8f __attribute__((ext_vector_type(8)));

// ---------------------------------------------------------------------------
// Kernel 1: zero the whole output (the bandwidth-critical part: ~134 MB).
// float4 grid-stride stores -> global_store_b128, pure store bandwidth.
// ---------------------------------------------------------------------------
__global__ void SpectralGate1D_zero_kernel(float4* __restrict__ out, long n4) {
    long i      = (long)blockIdx.x * blockDim.x + threadIdx.x;
    long stride = (long)gridDim.x * blockDim.x;
    float4 z; z.x = 0.f; z.y = 0.f; z.z = 0.f; z.w = 0.f;
    for (; i < n4; i += stride) out[i] = z;
}

// ---------------------------------------------------------------------------
// Kernel 2: spectral gate on the 64-step tail via per-d circulant matmul.
//   y[b,t,d] = sum_k x[b,TAIL_T0+k,d] * c_d[(t-k) & 63]
//   c_d[m]   = (1/64)(g0 + 2*sum_{f=1..31} g_f cos(2*pi*f*m/64) + g32*(-1)^m)
// One wave per d; Y_d(64x8) = C_d(64x64) @ W_d(64x8) with V_WMMA_F32_16X16X4_F32.
// ---------------------------------------------------------------------------
__global__ __launch_bounds__(256)
void SpectralGate1D_wmma_kernel(const float* __restrict__ x,
                                const float* __restrict__ mask,
                                const float* __restrict__ mix_w,
                                float* __restrict__ out) {
    __shared__ float s_g[DPB * NFREQ];            // sigmoid gates
    __shared__ float s_ctab[K_FFT];               // cos(2*pi*i/64) table
    __shared__ float s_c[DPB * K_FFT];            // circulant kernels c_d[m]
    __shared__ float s_w[DPB * K_FFT * B_DIM];    // window  [j][k][b]

    const int tid = threadIdx.x;
    const int d0  = blockIdx.x * DPB;

    // --- stage the tail window: x[b, TAIL_T0+k, d0+j] -> s_w[j][k][b]
    // idx = ((b*64 + k)*8 + j): consecutive tids hit consecutive d (32B chunks)
    for (int idx = tid; idx < DPB * K_FFT * B_DIM; idx += 256) {
        int j  = idx & (DPB - 1);
        int t2 = idx >> 3;               // b*64 + k
        int k  = t2 & (K_FFT - 1);
        int b  = t2 >> 6;
        s_w[j * (K_FFT * B_DIM) + k * B_DIM + b] =
            x[((long)b * T_DIM + TAIL_T0 + k) * (long)D_DIM + d0 + j];
    }
    // --- gates: g = sigmoid(mask[d, f])
    if (tid < DPB * NFREQ) {
        int j = tid / NFREQ, f = tid % NFREQ;
        float m = mask[(long)(d0 + j) * NFREQ + f];
        s_g[tid] = 1.0f / (1.0f + __expf(-m));
    }
    // --- cos table: cos(2*pi*i/64) = cos(i * pi/32)
    if (tid < K_FFT) s_ctab[tid] = __cosf((float)tid * 0.09817477042468103f);
    __syncthreads();

    // --- circulant kernel c_d[m] (uses periodicity: cos idx = (f*m) & 63)
    for (int idx = tid; idx < DPB * K_FFT; idx += 256) {
        int j = idx >> 6, m = idx & 63;
        const float* g = &s_g[j * NFREQ];
        float c = g[0] + ((m & 1) ? -g[NFREQ - 1] : g[NFREQ - 1]);
#pragma unroll
        for (int f = 1; f < NFREQ - 1; ++f)
            c += 2.0f * g[f] * s_ctab[(f * m) & 63];
        s_c[idx] = c * (1.0f / (float)K_FFT);
    }
    __syncthreads();

    // --- per-wave WMMA: Y_d = C_d @ W_d, M=64 (4 tiles) x N=16 (8 valid) x K=64
    const int wave = tid >> 5;
    const int lane = tid & 31;
    const int d    = d0 + wave;
    const float* cw = &s_c[wave * K_FFT];
    const float* ww = &s_w[wave * (K_FFT * B_DIM)];

    const int n  = lane & 15;     // M-within-tile (A) / N column (B, C/D)
    const int hi = lane >> 4;     // 0 -> K pair (k0,k0+1), 1 -> (k0+2,k0+3)

    v8f acc[4] = {};              // 4 M-tiles x (16x16 f32 C/D = 8 VGPRs)

#pragma unroll 4
    for (int k0 = 0; k0 < K_FFT; k0 += 4) {
        const int kk = k0 + hi * 2;
        // B fragment (4x16, K x N): row-striped across lanes
        v2f bf;
        bf.x = (n < B_DIM) ? ww[kk * B_DIM + n]       : 0.0f;
        bf.y = (n < B_DIM) ? ww[(kk + 1) * B_DIM + n] : 0.0f;
#pragma unroll
        for (int mt = 0; mt < 4; ++mt) {
            const int m = mt * 16 + n;          // global row t of this tile
            // A fragment (16x4, M x K): C_d[m][k] = c_d[(m-k) & 63]
            v2f af;
            af.x = cw[(m - kk) & 63];
            af.y = cw[(m - kk - 1) & 63];
            acc[mt] = __builtin_amdgcn_wmma_f32_16x16x4_f32(
                false, af, false, bf, (short)0, acc[mt], false, false);
        }
    }

    // --- store: C/D layout: VGPR r, lanes0-15 -> M = mt*16 + r, lanes16-31 -> +8
    if (n < B_DIM) {
        const float mw = mix_w[d];   // * GATE_STRENGTH (== 1.0f)
#pragma unroll
        for (int mt = 0; mt < 4; ++mt) {
#pragma unroll
            for (int r = 0; r < 8; ++r) {
                int t = mt * 16 + hi * 8 + r;
                out[((long)n * T_DIM + TAIL_T0 + t) * (long)D_DIM + d] =
                    acc[mt][r] * mw;
            }
        }
    }
}

// ---------------------------------------------------------------------------
extern "C" void kernel_launch(void* const* d_in, const int* in_sizes, int n_in,
                              void* d_out, int out_size, void* d_ws, size_t ws_size,
                              hipStream_t stream) {
    const float* x     = (const float*)d_in[0];
    const float* mask  = (const float*)d_in[1];
    const float* mix_w = (const float*)d_in[2];
    float* out = (float*)d_out;

    // 1) zero-fill the output (dominant cost: ~134 MB of stores)
    long n4 = (long)out_size / 4;   // out_size = 8*4096*1024, divisible by 4
    SpectralGate1D_zero_kernel<<<4096, 256, 0, stream>>>((float4*)out, n4);

    // 2) spectral gate on the 64-step tail (overwrites tail region)
    SpectralGate1D_wmma_kernel<<<D_DIM / DPB, 256, 0, stream>>>(x, mask, mix_w, out);
}